// N2_7653631721983
// MI455X (gfx1250) — compile-verified
//
#include <hip/hip_runtime.h>
#include <math.h>

typedef __attribute__((ext_vector_type(2))) float v2f;
typedef __attribute__((ext_vector_type(8))) float v8f;

#define NJ 24
#define NK 25  // N+1 output steps

// ---------------------------------------------------------------------------
// Kernel 1: build the 25 Rodrigues 4x4 matrices WM[k] into workspace.
// WM[k] = [[R | t],[0 0 0 1]],  R = K*sin(a) + (I3 - uu^T)*cos(a) + uu^T
// k < 24 -> per-joint weights, k == 24 -> end-effector weights.
// ---------------------------------------------------------------------------
__global__ void build_wm_kernel(const float* __restrict__ aW,
                                const float* __restrict__ uW,
                                const float* __restrict__ tW,
                                const float* __restrict__ eaW,
                                const float* __restrict__ euW,
                                const float* __restrict__ etW,
                                float* __restrict__ wm) {
  const int k = threadIdx.x;
  if (k >= NK) return;
  float a, ux, uy, uz, tx, ty, tz;
  if (k < NJ) {
    a  = aW[k];
    ux = uW[3*k+0]; uy = uW[3*k+1]; uz = uW[3*k+2];
    tx = tW[3*k+0]; ty = tW[3*k+1]; tz = tW[3*k+2];
  } else {
    a  = eaW[0];
    ux = euW[0]; uy = euW[1]; uz = euW[2];
    tx = etW[0]; ty = etW[1]; tz = etW[2];
  }
  // sigmoid then normalize
  ux = 1.0f / (1.0f + expf(-ux));
  uy = 1.0f / (1.0f + expf(-uy));
  uz = 1.0f / (1.0f + expf(-uz));
  const float inv = 1.0f / sqrtf(ux*ux + uy*uy + uz*uz);
  ux *= inv; uy *= inv; uz *= inv;
  const float sa = sinf(a), ca = cosf(a);
  const float o00 = ux*ux, o01 = ux*uy, o02 = ux*uz;
  const float o11 = uy*uy, o12 = uy*uz, o22 = uz*uz;
  float* w = wm + k*16;
  w[0]  = (1.0f - o00)*ca + o00;
  w[1]  = (-uz)*sa - o01*ca + o01;
  w[2]  = ( uy)*sa - o02*ca + o02;
  w[3]  = tx;
  w[4]  = ( uz)*sa - o01*ca + o01;
  w[5]  = (1.0f - o11)*ca + o11;
  w[6]  = (-ux)*sa - o12*ca + o12;
  w[7]  = ty;
  w[8]  = (-uy)*sa - o02*ca + o02;
  w[9]  = ( ux)*sa - o12*ca + o12;
  w[10] = (1.0f - o22)*ca + o22;
  w[11] = tz;
  w[12] = 0.0f; w[13] = 0.0f; w[14] = 0.0f; w[15] = 1.0f;
}

// ---------------------------------------------------------------------------
// Kernel 2 (main): 1 thread = 1 batch element for the sequential 4x4 scan,
// wave-cooperative WMMA for out[b,k] = cum[b,k] @ WM[k].
// A (16x4) = four elements' carry matrices stacked; B (4x16) = WM[k]
// replicated in each 4-column block -> every column block of D holds the
// per-element product, so D is stored directly (masked to one copy).
// ---------------------------------------------------------------------------
__global__ __launch_bounds__(256) void chain_wmma_kernel(
    const float* __restrict__ I, const float* __restrict__ wm,
    float* __restrict__ out) {
  __shared__ __align__(16) float wmLds[NK * 16];     // 1600 B
  __shared__ __align__(16) float stage[8][32 * 20];  // 20 KB, per-wave slots

  const int tid  = threadIdx.x;
  const int lane = tid & 31;
  const int wave = tid >> 5;
  const int b    = blockIdx.x * 256 + tid;

  // ---- async-copy WM (100 x 16B) into LDS; per-wave redundant copy so the
  // per-wave ASYNCcnt wait suffices (no block barrier, identical writes race
  // benignly).
  {
    const unsigned ldsBase = (unsigned)(uintptr_t)(&wmLds[0]);
#pragma unroll
    for (int i = 0; i < 4; ++i) {
      const int chunk = i * 32 + lane;
      if (chunk < 100) {
        const unsigned loff = ldsBase + (unsigned)chunk * 16u;
        const float* g = wm + chunk * 4;
        asm volatile("global_load_async_to_lds_b128 %0, %1, off"
                     :: "v"(loff), "v"(g) : "memory");
      }
    }
    asm volatile("s_wait_asynccnt 0x0" ::: "memory");
  }

  // ---- init carry = I[b, 0]
  const float4* Ib4 = reinterpret_cast<const float4*>(I) + (size_t)b * (NJ * 4);
  float c[16];
  {
    float4 t0 = Ib4[0], t1 = Ib4[1], t2 = Ib4[2], t3 = Ib4[3];
    c[0]=t0.x;  c[1]=t0.y;  c[2]=t0.z;  c[3]=t0.w;
    c[4]=t1.x;  c[5]=t1.y;  c[6]=t1.z;  c[7]=t1.w;
    c[8]=t2.x;  c[9]=t2.y;  c[10]=t2.z; c[11]=t2.w;
    c[12]=t3.x; c[13]=t3.y; c[14]=t3.z; c[15]=t3.w;
  }

  const int  h      = lane >> 4;          // K-half (A: K 0,1 / 2,3; B: rows 2h,2h+1)
  const int  cB     = lane & 3;           // column within a 4-col block
  const int  mA     = (lane & 15) >> 2;   // element within group of 4
  const int  rA     = lane & 3;           // row within the 4x4
  const bool active = ((lane & 12) == 0); // lanes 0-3 and 16-19 store results
  float* sl = &stage[wave][lane * 20];    // this thread's staging slot
  const float* sw = &stage[wave][0];

#pragma unroll 1
  for (int k = 0; k < NK; ++k) {
    // chain update: carry = carry @ I[b,k]   (k = 1..23)
    if (k >= 1 && k < NJ) {
      float M[16];
      {
        float4 m0 = Ib4[k*4+0], m1 = Ib4[k*4+1], m2 = Ib4[k*4+2], m3 = Ib4[k*4+3];
        M[0]=m0.x;  M[1]=m0.y;  M[2]=m0.z;  M[3]=m0.w;
        M[4]=m1.x;  M[5]=m1.y;  M[6]=m1.z;  M[7]=m1.w;
        M[8]=m2.x;  M[9]=m2.y;  M[10]=m2.z; M[11]=m2.w;
        M[12]=m3.x; M[13]=m3.y; M[14]=m3.z; M[15]=m3.w;
      }
      float nc[16];
#pragma unroll
      for (int r = 0; r < 4; ++r) {
#pragma unroll
        for (int j = 0; j < 4; ++j) {
          nc[r*4+j] = fmaf(c[r*4+0], M[0*4+j],
                      fmaf(c[r*4+1], M[1*4+j],
                      fmaf(c[r*4+2], M[2*4+j],
                           c[r*4+3] * M[3*4+j])));
        }
      }
#pragma unroll
      for (int i = 0; i < 16; ++i) c[i] = nc[i];
    }

    // stage carry into LDS (same-wave LDS ops are processed in order; the asm
    // barriers only stop the compiler from reordering around the shared array)
    asm volatile("" ::: "memory");
#pragma unroll
    for (int i = 0; i < 16; ++i) sl[i] = c[i];
    asm volatile("" ::: "memory");

    // B operand: WM[k] replicated across all four 4-column blocks.
    // Guessed f32 4x16 B layout (mirrors A/C-D striping): VGPR0 = row 2h,
    // VGPR1 = row 2h+1, lane = column.
    v2f Bv;
    Bv.x = wmLds[k*16 + 8*h + cB];
    Bv.y = wmLds[k*16 + 8*h + 4 + cB];

#pragma unroll
    for (int g = 0; g < 8; ++g) {
      // A operand (16x4): row L = carry of element g*4 + (L>>2), row L&3;
      // VGPR0/1 hold K = 2h / 2h+1.
      const float* ap = sw + (g*4 + mA)*20 + rA*4 + 2*h;
      v2f Av = *reinterpret_cast<const v2f*>(ap);
      v8f zeroC = {0.f, 0.f, 0.f, 0.f, 0.f, 0.f, 0.f, 0.f};
      v8f d = __builtin_amdgcn_wmma_f32_16x16x4_f32(false, Av, false, Bv,
                                                    (short)0, zeroC,
                                                    false, false);
      // D[row,col]: VGPR vg holds row vg (lanes 0-15) / vg+8 (lanes 16-31),
      // lane&15 = col.  Keep only column block n==0 (lanes 0-3 / 16-19).
      if (active) {
        const int bg = blockIdx.x * 256 + wave * 32 + g * 4;
        const size_t base = ((size_t)bg * 25 + (size_t)k) * 16 +
                            (size_t)cB + (size_t)(2 * h) * 400;
#pragma unroll
        for (int vg = 0; vg < 8; ++vg) {
          __builtin_nontemporal_store(
              d[vg], &out[base + (size_t)((vg >> 2) * 400 + (vg & 3) * 4)]);
        }
      }
    }
  }
}

// ---------------------------------------------------------------------------
// Scalar tail for B % 256 != 0 (keeps WMMA waves fully populated). Dead code
// for the reference shapes (B = 65536) but correct in general.
// ---------------------------------------------------------------------------
__global__ void chain_tail_kernel(const float* __restrict__ I,
                                  const float* __restrict__ wm,
                                  float* __restrict__ out, int b0, int B) {
  const int b = b0 + blockIdx.x * 64 + threadIdx.x;
  if (b >= B) return;
  const float* Ib = I + (size_t)b * (NJ * 16);
  float c[16];
  for (int i = 0; i < 16; ++i) c[i] = Ib[i];
  for (int k = 0; k < NK; ++k) {
    if (k >= 1 && k < NJ) {
      const float* M = Ib + k * 16;
      float nc[16];
      for (int r = 0; r < 4; ++r)
        for (int j = 0; j < 4; ++j)
          nc[r*4+j] = c[r*4+0]*M[j] + c[r*4+1]*M[4+j] +
                      c[r*4+2]*M[8+j] + c[r*4+3]*M[12+j];
      for (int i = 0; i < 16; ++i) c[i] = nc[i];
    }
    const float* W = wm + k * 16;
    float* o = out + ((size_t)b * 25 + k) * 16;
    for (int r = 0; r < 4; ++r)
      for (int j = 0; j < 4; ++j)
        o[r*4+j] = c[r*4+0]*W[j] + c[r*4+1]*W[4+j] +
                   c[r*4+2]*W[8+j] + c[r*4+3]*W[12+j];
  }
}

// ---------------------------------------------------------------------------
extern "C" void kernel_launch(void* const* d_in, const int* in_sizes, int n_in,
                              void* d_out, int out_size, void* d_ws,
                              size_t ws_size, hipStream_t stream) {
  const float* I   = (const float*)d_in[0];
  const float* aW  = (const float*)d_in[1];
  const float* uW  = (const float*)d_in[2];
  const float* tW  = (const float*)d_in[3];
  const float* eaW = (const float*)d_in[4];
  const float* euW = (const float*)d_in[5];
  const float* etW = (const float*)d_in[6];
  float* out = (float*)d_out;
  float* wm  = (float*)d_ws;  // 25*16 floats = 1600 B of scratch

  const int B = in_sizes[0] / (NJ * 16);

  build_wm_kernel<<<1, 32, 0, stream>>>(aW, uW, tW, eaW, euW, etW, wm);

  const int Bmain = B & ~255;
  if (Bmain > 0) {
    chain_wmma_kernel<<<Bmain / 256, 256, 0, stream>>>(I, wm, out);
  }
  const int rem = B - Bmain;
  if (rem > 0) {
    chain_tail_kernel<<<(rem + 63) / 64, 64, 0, stream>>>(I, wm, out, Bmain, B);
  }
}